// MvnMixture_26456998544000
// MI455X (gfx1250) — compile-verified
//
#include <hip/hip_runtime.h>
#include <math.h>

typedef float v2f __attribute__((ext_vector_type(2)));
typedef float v8f __attribute__((ext_vector_type(8)));

#define BDIM 512     // D
#define KCOMP 16     // mixture components
#define RRANK 32     // low-rank r
#define NCOLS 544    // 512 V cols + 16 w2 cols + 16 w1 cols
#define AP 516       // padded LDS stride for A tile (bank-conflict free)
#define EP 548       // padded LDS stride for E tile

// ---------------------------------------------------------------------------
// Kernel A: per-component precompute.  grid=16 blocks (one per k), 256 thr.
//  - Bmat[n][k*32+r] = u/sigma ; Bmat[n][512+k] = mu/sigma ; Bmat[n][528+k]=1/sigma
//  - cap = I + (u/sigma)^T u ; L = chol(cap) ; Linv = L^{-1}
//  - c[k][r] = sum_n mu*u/sigma ; beta[k] = log w - 0.5*(const + logdet + c1)
// ---------------------------------------------------------------------------
__global__ __launch_bounds__(256)
void precompute_kernel(const float* __restrict__ mu, const float* __restrict__ sigma,
                       const float* __restrict__ u, const float* __restrict__ w,
                       float* __restrict__ Bmat, float* __restrict__ LinvG,
                       float* __restrict__ cG, float* __restrict__ betaG) {
    const int k = blockIdx.x;
    const int t = threadIdx.x;

    __shared__ float sig[BDIM];
    __shared__ float isig[BDIM];
    __shared__ float cap[32 * 33];
    __shared__ float linv[32 * 33];
    __shared__ float red[128];

    const float* sigk = sigma + k * BDIM;
    const float* muk  = mu + k * BDIM;
    const float* uk   = u + (size_t)k * BDIM * RRANK;

    for (int n = t; n < BDIM; n += 256) {
        float s = sigk[n];
        sig[n] = s;
        isig[n] = 1.0f / s;
    }
    __syncthreads();

    // Fill B-matrix columns owned by this component
    for (int idx = t; idx < BDIM * RRANK; idx += 256) {
        int n = idx >> 5, r = idx & 31;
        Bmat[n * NCOLS + k * 32 + r] = uk[n * 32 + r] * isig[n];
    }
    for (int n = t; n < BDIM; n += 256) {
        Bmat[n * NCOLS + 512 + k] = muk[n] * isig[n];
        Bmat[n * NCOLS + 528 + k] = isig[n];
    }

    // cap = I + u^T diag(1/sigma) u   (32x32)
    for (int idx = t; idx < 1024; idx += 256) {
        int r = idx >> 5, s = idx & 31;
        float acc = (r == s) ? 1.0f : 0.0f;
        for (int n = 0; n < BDIM; ++n)
            acc += uk[n * 32 + r] * uk[n * 32 + s] * isig[n];
        cap[r * 33 + s] = acc;
    }
    __syncthreads();

    // Right-looking Cholesky in LDS (lower triangle of cap becomes L)
    for (int j = 0; j < 32; ++j) {
        if (t == 0) cap[j * 33 + j] = sqrtf(cap[j * 33 + j]);
        __syncthreads();
        if (t > j && t < 32) cap[t * 33 + j] /= cap[j * 33 + j];
        __syncthreads();
        int wdt = 31 - j;
        for (int idx = t; idx < wdt * wdt; idx += 256) {
            int i = j + 1 + idx / wdt;
            int s = j + 1 + idx % wdt;
            cap[i * 33 + s] -= cap[i * 33 + j] * cap[s * 33 + j];
        }
        __syncthreads();
    }

    // Linv = L^{-1} via forward substitution, one column per thread
    for (int idx = t; idx < 32 * 33; idx += 256) linv[idx] = 0.0f;
    __syncthreads();
    if (t < 32) {
        int cc = t;
        for (int i = cc; i < 32; ++i) {
            float v;
            if (i == cc) {
                v = 1.0f / cap[i * 33 + i];
            } else {
                float s = 0.0f;
                for (int j2 = cc; j2 < i; ++j2)
                    s += cap[i * 33 + j2] * linv[j2 * 33 + cc];
                v = -s / cap[i * 33 + i];
            }
            linv[i * 33 + cc] = v;
        }
    }
    __syncthreads();
    for (int idx = t; idx < 1024; idx += 256) {
        int i = idx >> 5, j2 = idx & 31;
        LinvG[k * 1024 + idx] = linv[i * 33 + j2];
    }

    // c[k][r] = sum_n mu * u / sigma
    if (t < 32) {
        float acc = 0.0f;
        for (int n = 0; n < BDIM; ++n)
            acc += muk[n] * uk[n * 32 + t] * isig[n];
        cG[k * 32 + t] = acc;
    }

    // deterministic reductions: sum log sigma, c1 = sum mu^2/sigma
    if (t < 64) {
        float p1 = 0.0f, p2 = 0.0f;
        for (int n = t * 8; n < t * 8 + 8; ++n) {
            float m = muk[n];
            p1 += logf(sig[n]);
            p2 += m * m * isig[n];
        }
        red[t] = p1;
        red[64 + t] = p2;
    }
    __syncthreads();
    if (t == 0) {
        float ld2 = 0.0f;
        for (int j = 0; j < 32; ++j) ld2 += logf(cap[j * 33 + j]);
        float sl = 0.0f, c1 = 0.0f;
        for (int i = 0; i < 64; ++i) { sl += red[i]; c1 += red[64 + i]; }
        float logdet = 2.0f * ld2 + sl;
        const float CONSTT = 512.0f * 1.83787706640934548356f; // D*log(2*pi)
        betaG[k] = logf(w[k]) - 0.5f * (CONSTT + logdet + c1);
    }
}

// ---------------------------------------------------------------------------
// Kernel B: main GEMM (f32 WMMA) + fused epilogue.
// grid = 16384/32 = 512 blocks of 512 threads (16 waves: 2 M-waves x 8 N-waves)
// ---------------------------------------------------------------------------
__global__ __launch_bounds__(512)
void mvn_main_kernel(const float* __restrict__ point, const float* __restrict__ Bmat,
                     const float* __restrict__ LinvG, const float* __restrict__ cG,
                     const float* __restrict__ betaG, float* __restrict__ out) {
    __shared__ float As[32 * AP];     // 32 rows of point, padded
    __shared__ float Es[32 * EP];     // 32 x 544 GEMM result, padded
    __shared__ float cS[BDIM];        // c vector (512)
    __shared__ float red2[32 * 17];   // logsumexp staging

    const int t = threadIdx.x;
    const int b0 = blockIdx.x * 32;

    // Load 32x512 A tile (coalesced) and c vector
    for (int row = 0; row < 32; ++row)
        As[row * AP + t] = point[(size_t)(b0 + row) * BDIM + t];
    cS[t & 511] = cG[t & 511];
    __syncthreads();

    const int wid   = t >> 5;
    const int lane  = t & 31;
    const int waveM = wid >> 3;     // 0..1
    const int waveN = wid & 7;      // 0..7
    const int half  = lane >> 4;    // 0: K0/K1 lanes, 1: K2/K3 lanes
    const int ln    = lane & 15;
    const int m     = waveM * 16 + ln;   // A row for this lane
    const int rbase = waveM * 16 + half * 8;

    // Tiles 0..32: plain GEMM columns (V | w2). No select in the hot loop.
    for (int tile = waveN; tile < 33; tile += 8) {
        const int col0 = tile * 16;
        v8f acc = {0.f, 0.f, 0.f, 0.f, 0.f, 0.f, 0.f, 0.f};
        const float* bcol = Bmat + col0 + ln;
        for (int kk = 0; kk < BDIM; kk += 4) {
            const int ka = kk + 2 * half;
            v2f a = *(const v2f*)(&As[m * AP + ka]);
            v2f b;
            b.x = bcol[ka * NCOLS];
            b.y = bcol[(ka + 1) * NCOLS];
            acc = __builtin_amdgcn_wmma_f32_16x16x4_f32(
                false, a, false, b, (short)0, acc, false, false);
        }
        #pragma unroll
        for (int i = 0; i < 8; ++i)
            Es[(rbase + i) * EP + col0 + ln] = acc[i];
    }

    // Tile 33: p^2 @ (1/sigma) columns, squared-A variant.
    // Uniform per-wave branch => EXEC is all-ones inside for the WMMAs.
    if (waveN == 1) {
        const int col0 = 33 * 16;
        v8f acc = {0.f, 0.f, 0.f, 0.f, 0.f, 0.f, 0.f, 0.f};
        const float* bcol = Bmat + col0 + ln;
        for (int kk = 0; kk < BDIM; kk += 4) {
            const int ka = kk + 2 * half;
            v2f a = *(const v2f*)(&As[m * AP + ka]);
            a = a * a;
            v2f b;
            b.x = bcol[ka * NCOLS];
            b.y = bcol[(ka + 1) * NCOLS];
            acc = __builtin_amdgcn_wmma_f32_16x16x4_f32(
                false, a, false, b, (short)0, acc, false, false);
        }
        #pragma unroll
        for (int i = 0; i < 8; ++i)
            Es[(rbase + i) * EP + col0 + ln] = acc[i];
    }
    __syncthreads();

    // Epilogue: thread -> (b = t&31, k = t>>5); k uniform per wave
    {
        const int b = t & 31;
        const int k = t >> 5;
        const float* erow = &Es[b * EP];

        float d[32];
        #pragma unroll
        for (int r = 0; r < 32; ++r)
            d[r] = erow[k * 32 + r] - cS[k * 32 + r];

        // sum2 = d . (Linv_k d), Linv lower-triangular
        const float* lk = LinvG + k * 1024;
        float sum2 = 0.0f;
        #pragma unroll
        for (int i = 0; i < 32; ++i) {
            float y = 0.0f;
            #pragma unroll
            for (int j = 0; j <= i; ++j)
                y += lk[i * 32 + j] * d[j];
            sum2 += y * d[i];
        }

        const float q2 = erow[512 + k];   // p @ (mu/sigma)
        const float q1 = erow[528 + k];   // p^2 @ (1/sigma)
        const float lp = betaG[k] - 0.5f * (q1 - 2.0f * q2 - sum2);
        red2[b * 17 + k] = lp;
    }
    __syncthreads();

    // Deterministic logsumexp over 16 components per row
    if (t < 32) {
        float mx = red2[t * 17];
        for (int k2 = 1; k2 < 16; ++k2) mx = fmaxf(mx, red2[t * 17 + k2]);
        float s = 0.0f;
        for (int k2 = 0; k2 < 16; ++k2) s += expf(red2[t * 17 + k2] - mx);
        out[b0 + t] = mx + logf(s);
    }
}

// ---------------------------------------------------------------------------
extern "C" void kernel_launch(void* const* d_in, const int* in_sizes, int n_in,
                              void* d_out, int out_size, void* d_ws, size_t ws_size,
                              hipStream_t stream) {
    const float* point  = (const float*)d_in[0];
    const float* mu     = (const float*)d_in[1];
    const float* sigma  = (const float*)d_in[2];
    const float* u      = (const float*)d_in[3];
    const float* wts    = (const float*)d_in[4];

    float* ws    = (float*)d_ws;
    float* Bmat  = ws;                       // 512*544 floats
    float* Linv  = Bmat + BDIM * NCOLS;      // 16*32*32 floats
    float* cG    = Linv + KCOMP * RRANK * RRANK;  // 512 floats
    float* betaG = cG + KCOMP * RRANK;       // 16 floats

    precompute_kernel<<<KCOMP, 256, 0, stream>>>(mu, sigma, u, wts, Bmat, Linv, cG, betaG);
    mvn_main_kernel<<<16384 / 32, 512, 0, stream>>>(point, Bmat, Linv, cG, betaG, (float*)d_out);
}